// EquivariantAttention_87462714016125
// MI455X (gfx1250) — compile-verified
//
#include <hip/hip_runtime.h>
#include <hip/hip_bf16.h>
#include <math.h>

typedef float v2f __attribute__((ext_vector_type(2)));
typedef float v8f __attribute__((ext_vector_type(8)));

constexpr int Bn  = 4;
constexpr int Tn  = 512;
constexpr int En  = 256;
constexpr int Sn  = 768;   // T + E
constexpr int NH  = 16;
constexpr int HD  = 32;
constexpr int HID = 512;
constexpr int DH  = 96;    // 3*HD per head
constexpr int LGS = 772;   // padded logits stride
constexpr int KTS = 104;   // padded tile stride (16B-aligned rows)
constexpr int TILE_F = 16 * KTS;
constexpr float SCALING = 0.10206207261596577f; // sqrt(32/3)/32
constexpr double EPSD = 1e-5;

// ---------------------------------------------------------------------------
// Kernel 1: fused attention for one (b, h, 16-row query tile) per wave32.
// f32 WMMA (v_wmma_f32_16x16x4_f32) for QK^T and P·V.
// K/V tiles are gathered straight into LDS with GLOBAL_LOAD_ASYNC_TO_LDS_B128
// (ASYNCcnt-tracked) and double-buffered so the DMA for tile st+1 overlaps the
// WMMAs of tile st.
// ---------------------------------------------------------------------------
__global__ __launch_bounds__(32)
void attn_kernel(const float* __restrict__ qp, const float* __restrict__ kp,
                 const float* __restrict__ vp, const float* __restrict__ bias,
                 const unsigned char* __restrict__ kpm,
                 const long long* __restrict__ oci,
                 const float* __restrict__ law,
                 const unsigned char* __restrict__ em,
                 float* __restrict__ attn)
{
    // 16*772 logits + 2 double-buffered 16x104 tiles = 62,720 B
    __shared__ __align__(16) float smem[16 * LGS + 2 * TILE_F];
    float* lg  = smem;                 // logits / probs
    float* ktb = smem + 16 * LGS;      // tile buffers [2][16][104]
    float* qtp = ktb + TILE_F;         // Q tile overlays buffer 1

    const int bid  = blockIdx.x;
    const int tt   = bid & 31;         // query tile index (T/16 = 32)
    const int h    = (bid >> 5) & 15;  // head
    const int b    = bid >> 9;         // batch
    const int lane = threadIdx.x;
    const int m    = lane & 15;
    const int half = lane >> 4;
    const int t0   = tt * 16;

    // async gather of one 16x96 K/V tile into LDS buffer `buf` (12 x B128)
    auto issue_tile = [&](const float* __restrict__ basep, int s0, int buf) {
        int ss = s0 + m;
        int srcrow = (ss < Tn) ? ss : (int)oci[b * En + (ss - Tn)];
        // destination buffer may still have in-flight ds_load consumers
        asm volatile("s_wait_dscnt 0x0" ::: "memory");
#pragma unroll
        for (int it = 0; it < 12; ++it) {
            int f4  = lane + 32 * it;       // float4 index within tile
            int row = f4 / 24;              // 24 float4 per 96-float row
            int col = (f4 % 24) * 4;
            int src = __shfl(srcrow, row);
            int c = col >> 5, d0 = col & 31;
            const float* gp = basep + (((size_t)(b * Tn + src) * 3 + c) * HID) + h * HD + d0;
            unsigned lo = (unsigned)(size_t)(ktb + buf * TILE_F + row * KTS + col);
            asm volatile("global_load_async_to_lds_b128 %0, %1, off"
                         :: "v"(lo), "v"(gp) : "memory");
        }
    };

    // ---- stage scaled Q tile (16 x 96) into buffer 1, consume into frags ----
    for (int idx = lane; idx < 16 * DH; idx += 32) {
        int r = idx / DH, col = idx % DH;
        int c = col >> 5, d0 = col & 31;
        qtp[r * KTS + col] =
            qp[(((size_t)(b * Tn + t0 + r) * 3 + c) * HID) + h * HD + d0] * SCALING;
    }
    __syncthreads();

    // A-fragments of Q: element (M=lane&15, K=2*half+v)
    v2f qa[24];
    for (int kc = 0; kc < 24; ++kc) {
        qa[kc].x = qtp[m * KTS + kc * 4 + half * 2 + 0];
        qa[kc].y = qtp[m * KTS + kc * 4 + half * 2 + 1];
    }

    // ---- pass 1: logits = QK^T + bias, masked (double-buffered K tiles) ----
    issue_tile(kp, 0, 0);
    for (int st = 0; st < Sn / 16; ++st) {
        const int s0 = st * 16;
        if (st < Sn / 16 - 1) {
            issue_tile(kp, s0 + 16, (st + 1) & 1);
            asm volatile("s_wait_asynccnt 0xc" ::: "memory"); // current tile resident
        } else {
            asm volatile("s_wait_asynccnt 0x0" ::: "memory");
        }
        const float* kb_ = ktb + (st & 1) * TILE_F;

        v8f acc = {};
        for (int kc = 0; kc < 24; ++kc) {
            v2f bf;  // B element (K=dim=2*half+v, N=key=lane&15) = kh[key][dim]
            bf.x = kb_[m * KTS + kc * 4 + half * 2 + 0];
            bf.y = kb_[m * KTS + kc * 4 + half * 2 + 1];
            acc = __builtin_amdgcn_wmma_f32_16x16x4_f32(false, qa[kc], false, bf,
                                                        (short)0, acc, false, false);
        }
        // D layout: acc[i] -> (row = i + 8*half, col = lane&15)
        for (int i = 0; i < 8; ++i) {
            int row = i + half * 8;
            int t = t0 + row;
            int s = s0 + m;
            bool msk = (s < Tn) ? (kpm[b * Tn + s] != 0) : (em[b * En + (s - Tn)] != 0);
            float lw = law[((size_t)(b * Tn + t)) * Sn + s];
            float val;
            if (msk || lw <= 1e-5f) {
                val = -INFINITY;
            } else {
                val = acc[i] + bias[(((size_t)(b * NH + h) * Tn + t)) * Sn + s];
            }
            lg[row * LGS + s] = val;
        }
    }
    __syncthreads();

    // ---- softmax per row, then multiply by local_attention_weight ----
    for (int r = 0; r < 16; ++r) {
        float vals[24];
        float mx = -INFINITY;
        for (int j = 0; j < 24; ++j) {
            vals[j] = lg[r * LGS + lane + 32 * j];
            mx = fmaxf(mx, vals[j]);
        }
        for (int o = 16; o; o >>= 1) mx = fmaxf(mx, __shfl_xor(mx, o, 32));
        float sum = 0.f;
        for (int j = 0; j < 24; ++j) {
            float e = expf(vals[j] - mx);
            vals[j] = e;
            sum += e;
        }
        for (int o = 16; o; o >>= 1) sum += __shfl_xor(sum, o, 32);
        float inv = 1.0f / sum;
        const float* lrow = law + ((size_t)(b * Tn + t0 + r)) * Sn;
        for (int j = 0; j < 24; ++j) {
            int s = lane + 32 * j;
            lg[r * LGS + s] = vals[j] * inv * lrow[s];
        }
    }
    __syncthreads();

    // ---- pass 2: O = P @ V  (16 x 96), double-buffered V tiles ----
    v8f oacc[6] = {};
    issue_tile(vp, 0, 0);
    for (int st = 0; st < Sn / 16; ++st) {
        const int s0 = st * 16;
        if (st < Sn / 16 - 1) {
            issue_tile(vp, s0 + 16, (st + 1) & 1);
            asm volatile("s_wait_asynccnt 0xc" ::: "memory");
        } else {
            asm volatile("s_wait_asynccnt 0x0" ::: "memory");
        }
        const float* vb_ = ktb + (st & 1) * TILE_F;

        for (int kc = 0; kc < 4; ++kc) {
            v2f pa;  // A element (M=query=lane&15, K=key-in-tile=2*half+v)
            pa.x = lg[m * LGS + s0 + kc * 4 + half * 2 + 0];
            pa.y = lg[m * LGS + s0 + kc * 4 + half * 2 + 1];
            for (int j = 0; j < 6; ++j) {
                v2f vb;  // B element (K=key-in-tile, N=dim=j*16+lane&15)
                vb.x = vb_[(kc * 4 + half * 2 + 0) * KTS + j * 16 + m];
                vb.y = vb_[(kc * 4 + half * 2 + 1) * KTS + j * 16 + m];
                oacc[j] = __builtin_amdgcn_wmma_f32_16x16x4_f32(false, pa, false, vb,
                                                                (short)0, oacc[j], false, false);
            }
        }
    }

    // ---- epilogue: scatter back into (B, T, 3, HID) layout ----
    for (int j = 0; j < 6; ++j) {
        for (int i = 0; i < 8; ++i) {
            int row = i + half * 8;
            int d = j * 16 + m;
            int c = d >> 5, d0 = d & 31;
            attn[(((size_t)(b * Tn + t0 + row) * 3 + c) * HID) + h * HD + d0] = oacc[j][i];
        }
    }
}

// ---------------------------------------------------------------------------
// Kernel 2: equivariant layer norm, one wave32 per token, fp64 math.
// Closed-form symmetric 3x3 eigensolve + Lagrange spectral interpolation
// for inv_sqrt(covar). Rewrites the buffer in place (Y overwrites attn).
// ---------------------------------------------------------------------------
__device__ __forceinline__ double wred(double x) {
    for (int o = 16; o; o >>= 1) x += __shfl_xor(x, o, 32);
    return x;
}

__global__ __launch_bounds__(32)
void eqln_kernel(float* __restrict__ attn, const float* __restrict__ lnw)
{
    const int tok  = blockIdx.x;           // b*T + t
    const int lane = threadIdx.x;
    float* x = attn + (size_t)tok * 3 * HID;

    double s0 = 0, s1 = 0, s2 = 0;
    double p00 = 0, p01 = 0, p02 = 0, p11 = 0, p12 = 0, p22 = 0;
    for (int j = 0; j < HID / 32; ++j) {
        int i = lane + 32 * j;
        double a = (double)x[0 * HID + i];
        double bb = (double)x[1 * HID + i];
        double cc = (double)x[2 * HID + i];
        s0 += a; s1 += bb; s2 += cc;
        p00 += a * a; p01 += a * bb; p02 += a * cc;
        p11 += bb * bb; p12 += bb * cc; p22 += cc * cc;
    }
    s0 = wred(s0); s1 = wred(s1); s2 = wred(s2);
    p00 = wred(p00); p01 = wred(p01); p02 = wred(p02);
    p11 = wred(p11); p12 = wred(p12); p22 = wred(p22);

    const double invH = 1.0 / (double)HID;
    double m0 = s0 * invH, m1 = s1 * invH, m2 = s2 * invH;
    // covar = E[x x^T] - m m^T + EPS*diag(1,2,3)
    double c00 = p00 * invH - m0 * m0 + EPSD * 1.0;
    double c11 = p11 * invH - m1 * m1 + EPSD * 2.0;
    double c22 = p22 * invH - m2 * m2 + EPSD * 3.0;
    double c01 = p01 * invH - m0 * m1;
    double c02 = p02 * invH - m0 * m2;
    double c12 = p12 * invH - m1 * m2;

    // analytic eigenvalues of symmetric 3x3
    double qm = (c00 + c11 + c22) / 3.0;
    double pp1 = c01 * c01 + c02 * c02 + c12 * c12;
    double pp2 = (c00 - qm) * (c00 - qm) + (c11 - qm) * (c11 - qm) +
                 (c22 - qm) * (c22 - qm) + 2.0 * pp1;
    double p = sqrt(pp2 / 6.0);

    double M00, M01, M02, M11, M12, M22;
    bool iso = (p < 1e-30);
    double l0 = qm, l1 = qm, l2 = qm;
    if (!iso) {
        double ip = 1.0 / p;
        double b00 = (c00 - qm) * ip, b11 = (c11 - qm) * ip, b22 = (c22 - qm) * ip;
        double b01 = c01 * ip, b02 = c02 * ip, b12 = c12 * ip;
        double detB = b00 * (b11 * b22 - b12 * b12) -
                      b01 * (b01 * b22 - b12 * b02) +
                      b02 * (b01 * b12 - b11 * b02);
        double r = 0.5 * detB;
        r = fmin(1.0, fmax(-1.0, r));
        double phi = acos(r) / 3.0;
        l0 = qm + 2.0 * p * cos(phi);
        l2 = qm + 2.0 * p * cos(phi + 2.0943951023931953); // +2pi/3 (smallest)
        l1 = 3.0 * qm - l0 - l2;
    }
    double d0 = (l0 - l1) * (l0 - l2);
    double d1 = (l1 - l0) * (l1 - l2);
    double d2 = (l2 - l0) * (l2 - l1);
    if (iso || fabs(d0) < 1e-24 || fabs(d1) < 1e-24 || fabs(d2) < 1e-24) {
        double f = 1.0 / sqrt(qm + EPSD);
        M00 = f; M11 = f; M22 = f; M01 = 0; M02 = 0; M12 = 0;
    } else {
        // A^2 (symmetric)
        double a200 = c00 * c00 + c01 * c01 + c02 * c02;
        double a201 = c00 * c01 + c01 * c11 + c02 * c12;
        double a202 = c00 * c02 + c01 * c12 + c02 * c22;
        double a211 = c01 * c01 + c11 * c11 + c12 * c12;
        double a212 = c01 * c02 + c11 * c12 + c12 * c22;
        double a222 = c02 * c02 + c12 * c12 + c22 * c22;
        double f0 = 1.0 / sqrt(l0 + EPSD);
        double f1 = 1.0 / sqrt(l1 + EPSD);
        double f2 = 1.0 / sqrt(l2 + EPSD);
        double k0 = f0 / d0, k1 = f1 / d1, k2 = f2 / d2;
        // f(A) = sum_i k_i * (A^2 - (lj+lk) A + lj*lk I)
        double u0 = l1 + l2, w0 = l1 * l2;
        double u1 = l0 + l2, w1 = l0 * l2;
        double u2 = l0 + l1, w2 = l0 * l1;
        M00 = k0 * (a200 - u0 * c00 + w0) + k1 * (a200 - u1 * c00 + w1) + k2 * (a200 - u2 * c00 + w2);
        M11 = k0 * (a211 - u0 * c11 + w0) + k1 * (a211 - u1 * c11 + w1) + k2 * (a211 - u2 * c11 + w2);
        M22 = k0 * (a222 - u0 * c22 + w0) + k1 * (a222 - u1 * c22 + w1) + k2 * (a222 - u2 * c22 + w2);
        M01 = k0 * (a201 - u0 * c01) + k1 * (a201 - u1 * c01) + k2 * (a201 - u2 * c01);
        M02 = k0 * (a202 - u0 * c02) + k1 * (a202 - u1 * c02) + k2 * (a202 - u2 * c02);
        M12 = k0 * (a212 - u0 * c12) + k1 * (a212 - u1 * c12) + k2 * (a212 - u2 * c12);
    }

    // pass 2: y = M @ (x - mean), * ln_weight, write back in place
    for (int j = 0; j < HID / 32; ++j) {
        int i = lane + 32 * j;
        double x0 = (double)x[0 * HID + i] - m0;
        double x1 = (double)x[1 * HID + i] - m1;
        double x2 = (double)x[2 * HID + i] - m2;
        float w = lnw[i];
        x[0 * HID + i] = (float)(M00 * x0 + M01 * x1 + M02 * x2) * w;
        x[1 * HID + i] = (float)(M01 * x0 + M11 * x1 + M12 * x2) * w;
        x[2 * HID + i] = (float)(M02 * x0 + M12 * x1 + M22 * x2) * w;
    }
}

// ---------------------------------------------------------------------------
// Kernel 3: out = Y @ w_out^T  as a (6144 x 512) x (512 x 512) f32 WMMA GEMM.
// One wave per 16x64 output tile; Y-tile + W k-slabs staged in LDS.
// ---------------------------------------------------------------------------
__global__ __launch_bounds__(32)
void proj_kernel(const float* __restrict__ Y, const float* __restrict__ w,
                 float* __restrict__ out)
{
    __shared__ float yl[16][514];  // 32,896 B
    __shared__ float wl[64][68];   // 17,408 B

    const int lane = threadIdx.x;
    const int m = lane & 15, half = lane >> 4;
    const int m0 = blockIdx.x * 16;   // 384 tiles
    const int n0 = blockIdx.y * 64;   // 8 tiles

    for (int idx = lane; idx < 16 * 512; idx += 32) {
        int r = idx >> 9, col = idx & 511;
        yl[r][col] = Y[(size_t)(m0 + r) * HID + col];
    }

    v8f acc[4] = {};
    for (int kb = 0; kb < 8; ++kb) {
        const int k0 = kb * 64;
        __syncthreads();
        for (int idx = lane; idx < 64 * 64; idx += 32) {
            int r = idx >> 6, col = idx & 63;
            wl[r][col] = w[(size_t)(n0 + r) * HID + k0 + col];
        }
        __syncthreads();
        for (int kc = 0; kc < 16; ++kc) {
            v2f a;  // A element (M=lane&15, K=2*half+v)
            a.x = yl[m][k0 + kc * 4 + half * 2 + 0];
            a.y = yl[m][k0 + kc * 4 + half * 2 + 1];
            for (int j = 0; j < 4; ++j) {
                v2f bf;  // B element (K, N=j*16+lane&15) = w[n][k]
                bf.x = wl[j * 16 + m][kc * 4 + half * 2 + 0];
                bf.y = wl[j * 16 + m][kc * 4 + half * 2 + 1];
                acc[j] = __builtin_amdgcn_wmma_f32_16x16x4_f32(false, a, false, bf,
                                                               (short)0, acc[j], false, false);
            }
        }
    }

    for (int j = 0; j < 4; ++j) {
        for (int i = 0; i < 8; ++i) {
            int row = m0 + i + half * 8;
            int col = n0 + j * 16 + m;
            out[(size_t)row * HID + col] = acc[j][i];
        }
    }
}

// ---------------------------------------------------------------------------
extern "C" void kernel_launch(void* const* d_in, const int* in_sizes, int n_in,
                              void* d_out, int out_size, void* d_ws, size_t ws_size,
                              hipStream_t stream)
{
    const float*         q    = (const float*)d_in[0];
    const float*         k    = (const float*)d_in[1];
    const float*         v    = (const float*)d_in[2];
    const float*         bias = (const float*)d_in[3];
    const unsigned char* kpm  = (const unsigned char*)d_in[4]; // bool (B,T)
    const long long*     oci  = (const long long*)d_in[5];     // int64 (B,E)
    const float*         law  = (const float*)d_in[6];
    const unsigned char* em   = (const unsigned char*)d_in[7]; // bool (B,E)
    const float*         wout = (const float*)d_in[8];
    const float*         lnw  = (const float*)d_in[9];
    float*               out  = (float*)d_out;
    float*               attn = (float*)d_ws;  // B*T*3*HID floats = 12.6 MB

    // 1) attention: one wave per (b, h, 16-query tile)
    attn_kernel<<<dim3(Bn * NH * (Tn / 16)), dim3(32), 0, stream>>>(
        q, k, v, bias, kpm, oci, law, em, attn);

    // 2) equivariant layer norm in place: one wave per token
    eqln_kernel<<<dim3(Bn * Tn), dim3(32), 0, stream>>>(attn, lnw);

    // 3) projection GEMM: (B*T*3, HID) @ w_out^T
    proj_kernel<<<dim3((Bn * Tn * 3) / 16, HID / 64), dim3(32), 0, stream>>>(
        attn, wout, out);
}